// HMM_39384850104588
// MI455X (gfx1250) — compile-verified
//
#include <hip/hip_runtime.h>
#include <math.h>

typedef __attribute__((ext_vector_type(2))) float v2f;
typedef __attribute__((ext_vector_type(8))) float v8f;

#define N_STATES 64
#define NF 39
#define KAUG 80              // 2*39 padded to multiple of 4
#define KSTRIDE 82           // padded LDS row stride: bank-conflict-free, 8B aligned
#define TSTEPS 200000
#define NEG (-1.0e30f)
#define LOG2PI 1.8378770664093453f
#define WAVES_PER_BLOCK 8
#define ROWS_PER_WAVE 16

// ---------------------------------------------------------------------------
// Emission kernel: ll[t,n] = sum_k aug[t,k] * W[k,n] + const[n]
//   aug[t, k<39]  = x[t,k]          W[k<39 , n] = mu[n,k] * exp(-lv[n,k])
//   aug[t, 39..77]= x[t,k-39]^2     W[39..77,n] = -0.5   * exp(-lv[n,k-39])
// Written to d_out row (t+1); columns 0 and 63 forced to NEG.
// Uses V_WMMA_F32_16X16X4_F32, one 16x64 output tile per wave.
// LDS layouts are [row][k] with padded stride so every WMMA fragment is a
// contiguous 8-byte ds_load_b64 straight into an even-aligned VGPR pair.
// ---------------------------------------------------------------------------
__global__ __launch_bounds__(256) void HMM_emis_kernel(
    const float* __restrict__ data,
    const float* __restrict__ mu,
    const float* __restrict__ log_var,
    float* __restrict__ out) {
  __shared__ float sB[N_STATES * KSTRIDE];                       // [n][k]
  __shared__ float sC[N_STATES];
  __shared__ float sA[WAVES_PER_BLOCK][ROWS_PER_WAVE * KSTRIDE]; // [row][k]

  const int tid = threadIdx.x;

  // Stage B^T = W[n][k] in LDS (transposed so K-pairs are contiguous).
  for (int idx = tid; idx < N_STATES * KAUG; idx += blockDim.x) {
    int n = idx / KAUG;
    int k = idx % KAUG;
    float v = 0.0f;
    if (k < NF) {
      float iv = __expf(-log_var[n * NF + k]);
      v = mu[n * NF + k] * iv;
    } else if (k < 2 * NF) {
      int f = k - NF;
      float iv = __expf(-log_var[n * NF + f]);
      v = -0.5f * iv;
    }
    sB[n * KSTRIDE + k] = v;
  }
  // Per-state constant.
  if (tid < N_STATES) {
    float c = 0.0f;
    for (int f = 0; f < NF; ++f) {
      float lv = log_var[tid * NF + f];
      float iv = __expf(-lv);
      float m  = mu[tid * NF + f];
      c += m * m * iv + lv + LOG2PI;
    }
    sC[tid] = -0.5f * c;
  }

  const int wave = tid >> 5;
  const int lane = tid & 31;
  const int tile = blockIdx.x * WAVES_PER_BLOCK + wave;
  const int ntiles = TSTEPS / ROWS_PER_WAVE;

  // Stage this wave's 16x80 augmented-feature tile.
  if (tile < ntiles) {
    const int t0 = tile * ROWS_PER_WAVE;
    float* a = sA[wave];
    for (int idx = lane; idx < ROWS_PER_WAVE * KAUG; idx += 32) {
      int r = idx / KAUG;
      int k = idx % KAUG;
      float v = 0.0f;
      if (k < 2 * NF) {
        int f = (k < NF) ? k : (k - NF);
        float x = data[(size_t)(t0 + r) * NF + f];
        v = (k < NF) ? x : x * x;
      }
      a[r * KSTRIDE + k] = v;
    }
  }
  __syncthreads();

  if (tile < ntiles) {
    const int t0    = tile * ROWS_PER_WAVE;
    const float* a  = sA[wave];
    const int half  = lane >> 4;      // 0: lanes 0-15, 1: lanes 16-31
    const int half2 = half * 2;
    const int lid   = lane & 15;
    const float* arow = a + lid * KSTRIDE + half2;   // A row M=lid, K base
    for (int nt = 0; nt < 4; ++nt) {
      const int n0  = nt * 16;
      const int col = n0 + lid;
      const float* brow = sB + col * KSTRIDE + half2; // B col N=col, K base
      v8f acc = {};
#pragma unroll
      for (int kk = 0; kk < KAUG / 4; ++kk) {
        // A 16x4: lanes 0-15 -> K={k0,k0+1}; lanes 16-31 -> K={k0+2,k0+3}
        // B 4x16: same K split, N = n0+lid. Both contiguous 8B LDS loads.
        v2f av = *(const v2f*)(arow + kk * 4);
        v2f bv = *(const v2f*)(brow + kk * 4);
        acc = __builtin_amdgcn_wmma_f32_16x16x4_f32(
            false, av, false, bv, (short)0, acc, false, false);
      }
      const bool inner = (col > 0) && (col < N_STATES - 1);
      const float cadd = sC[col];
      const int rbase  = half * 8;    // D: VGPR r -> M = r + 8*half
#pragma unroll
      for (int r = 0; r < 8; ++r) {
        const int trow = t0 + rbase + r;
        float v = inner ? (acc[r] + cadd) : NEG;
        out[(size_t)(trow + 1) * N_STATES + col] = v;
      }
    }
  }
}

// ---------------------------------------------------------------------------
// Forward scan, in place over d_out rows 1..T. One wave; lane l owns states
// 2l and 2l+1. The chain transition matrix means each state has exactly two
// finite predecessors (self + left neighbor); dense logsumexp over the NEG
// entries underflows to 0 in fp32, so a 2-term lse is numerically identical.
// ---------------------------------------------------------------------------
__device__ __forceinline__ float lse2(float x, float y) {
  float m = fmaxf(x, y);
  float d = fminf(x, y) - m;            // <= 0, finite
  return m + __logf(1.0f + __expf(d));
}

__global__ void HMM_scan_kernel(const float* __restrict__ log_init,
                                const float* __restrict__ log_trans,
                                float* __restrict__ out) {
  const int lane = threadIdx.x;         // 0..31 (wave32)
  const int j0 = 2 * lane, j1 = 2 * lane + 1;

  const float ldiag0 = log_trans[j0 * N_STATES + j0];
  const float ldiag1 = log_trans[j1 * N_STATES + j1];
  const float lup0   = (j0 > 0) ? log_trans[(j0 - 1) * N_STATES + j0] : NEG;
  const float lup1   = log_trans[j0 * N_STATES + j1];

  float a0 = log_init[j0];
  float a1 = log_init[j1];

  // Row 0 = log_init.
  *(float2*)(out + j0) = make_float2(a0, a1);

  const int PD = 16;                    // prefetch depth (T % PD == 0)
  float2 eb[PD];
#pragma unroll
  for (int i = 0; i < PD; ++i)
    eb[i] = *(const float2*)(out + (size_t)(i + 1) * N_STATES + j0);

  int t = 0;
  for (; t < TSTEPS - PD; t += PD) {
#pragma unroll
    for (int i = 0; i < PD; ++i) {
      float2 e = eb[i];
      eb[i] = *(const float2*)(out + (size_t)(t + i + PD + 1) * N_STATES + j0);
      float am = __shfl_up(a1, 1, 32);  // alpha[2l-1] from lane l-1
      float c0 = lse2(am + lup0, a0 + ldiag0) + e.x;
      float c1 = lse2(a0 + lup1, a1 + ldiag1) + e.y;
      a0 = fmaxf(c0, NEG);
      a1 = fmaxf(c1, NEG);
      *(float2*)(out + (size_t)(t + i + 1) * N_STATES + j0) = make_float2(a0, a1);
    }
  }
  // Epilogue: last PD steps, no prefetch.
#pragma unroll
  for (int i = 0; i < PD; ++i) {
    float2 e = eb[i];
    float am = __shfl_up(a1, 1, 32);
    float c0 = lse2(am + lup0, a0 + ldiag0) + e.x;
    float c1 = lse2(a0 + lup1, a1 + ldiag1) + e.y;
    a0 = fmaxf(c0, NEG);
    a1 = fmaxf(c1, NEG);
    *(float2*)(out + (size_t)(t + i + 1) * N_STATES + j0) = make_float2(a0, a1);
  }

  // Row T+1 = zeros.
  *(float2*)(out + (size_t)(TSTEPS + 1) * N_STATES + j0) = make_float2(0.0f, 0.0f);
}

extern "C" void kernel_launch(void* const* d_in, const int* in_sizes, int n_in,
                              void* d_out, int out_size, void* d_ws, size_t ws_size,
                              hipStream_t stream) {
  (void)in_sizes; (void)n_in; (void)out_size; (void)d_ws; (void)ws_size;
  const float* data      = (const float*)d_in[0];
  const float* mu        = (const float*)d_in[1];
  const float* log_var   = (const float*)d_in[2];
  const float* log_trans = (const float*)d_in[3];
  const float* log_init  = (const float*)d_in[4];
  float* out = (float*)d_out;

  const int tiles  = TSTEPS / ROWS_PER_WAVE;                       // 12500
  const int blocks = (tiles + WAVES_PER_BLOCK - 1) / WAVES_PER_BLOCK;
  HMM_emis_kernel<<<blocks, 256, 0, stream>>>(data, mu, log_var, out);
  HMM_scan_kernel<<<1, 32, 0, stream>>>(log_init, log_trans, out);
}